// EquivBBDM_27693949125355
// MI455X (gfx1250) — compile-verified
//
#include <hip/hip_runtime.h>
#include <math.h>

typedef float v2f __attribute__((ext_vector_type(2)));
typedef float v8f __attribute__((ext_vector_type(8)));

#define NPG 256   // nodes per graph (reference: n=256)
#define DIM 32    // feature dim (reference: d=32, == wave32 width)

// ---------------- Kernel 1: per-graph frac transform + q_sample ----------------
__global__ void k_setup(const float* __restrict__ pos,
                        const float* __restrict__ pos_relaxed,
                        const float* __restrict__ cell,
                        const float* __restrict__ noise,
                        const float* __restrict__ m_sched,
                        const float* __restrict__ var_sched,
                        const int*   __restrict__ t_graph,
                        float* __restrict__ pt_ws,    // [N][4]
                        float* __restrict__ obj_ws)   // [N][4]
{
    const int g = blockIdx.x;
    const int j = threadIdx.x;
    const int i = g * NPG + j;

    const float* c = cell + g * 9;
    float c00=c[0],c01=c[1],c02=c[2],c10=c[3],c11=c[4],c12=c[5],c20=c[6],c21=c[7],c22=c[8];
    float det = c00*(c11*c22-c12*c21) - c01*(c10*c22-c12*c20) + c02*(c10*c21-c11*c20);
    float id = 1.0f / det;
    float i00 =  (c11*c22-c12*c21)*id, i01 = -(c01*c22-c02*c21)*id, i02 =  (c01*c12-c02*c11)*id;
    float i10 = -(c10*c22-c12*c20)*id, i11 =  (c00*c22-c02*c20)*id, i12 = -(c00*c12-c02*c10)*id;
    float i20 =  (c10*c21-c11*c20)*id, i21 = -(c00*c21-c01*c20)*id, i22 =  (c00*c11-c01*c10)*id;

    float px=pos[3*i],        py=pos[3*i+1],        pz=pos[3*i+2];
    float qx=pos_relaxed[3*i],qy=pos_relaxed[3*i+1],qz=pos_relaxed[3*i+2];
    float nx=noise[3*i],      ny=noise[3*i+1],      nz=noise[3*i+2];

    // row-vector times inv(cell):  f[c] = sum_r v[r] * inv[r][c]
    float pix = px*i00 + py*i10 + pz*i20, piy = px*i01 + py*i11 + pz*i21, piz = px*i02 + py*i12 + pz*i22;
    float prx = qx*i00 + qy*i10 + qz*i20, pry = qx*i01 + qy*i11 + qz*i21, prz = qx*i02 + qy*i12 + qz*i22;
    float nfx = nx*i00 + ny*i10 + nz*i20, nfy = nx*i01 + ny*i11 + nz*i21, nfz = nx*i02 + ny*i12 + nz*i22;

    int   t  = t_graph[g];
    float mt = m_sched[t];
    float sg = sqrtf(var_sched[t]);

    float dx = pix - prx, dy = piy - pry, dz = piz - prz;
    dx -= floorf(dx + 0.5f); dy -= floorf(dy + 0.5f); dz -= floorf(dz + 0.5f);
    float ox = mt*dx + sg*nfx, oy = mt*dy + sg*nfy, oz = mt*dz + sg*nfz;
    float tx = ox + prx, ty = oy + pry, tz = oz + prz;

    ((float4*)pt_ws)[i]  = make_float4(tx, ty, tz, 0.0f);
    ((float4*)obj_ws)[i] = make_float4(ox, oy, oz, 0.0f);
}

// ---------------- Kernel 2: h = emb_z[z] + emb_t[t] ----------------
__global__ void k_h(const float* __restrict__ emb_z,
                    const float* __restrict__ emb_t,
                    const int*   __restrict__ atomic_numbers,
                    const int*   __restrict__ t_graph,
                    float* __restrict__ h, int total)
{
    int idx = blockIdx.x * blockDim.x + threadIdx.x;
    if (idx >= total) return;
    int node = idx >> 5;
    int k    = idx & 31;
    int g    = node >> 8;                 // n = 256
    int z    = atomic_numbers[node];
    int t    = t_graph[g];
    h[idx] = emb_z[z*DIM + k] + emb_t[t*DIM + k];
}

// ---------------- Kernel 3: hW = h @ W_msg  via V_WMMA_F32_16X16X4_F32 ----------------
// One wave computes a 16x32 output tile: 2 accumulators, 8 K-steps of 4 -> 16 WMMAs.
__global__ void k_hW_wmma(const float* __restrict__ h,
                          const float* __restrict__ W,   // [32][32] row-major
                          float* __restrict__ hW)
{
    const int wave = threadIdx.x >> 5;
    const int lane = threadIdx.x & 31;
    const int tile = blockIdx.x * 8 + wave;
    const int rowBase = tile * 16;
    const int col  = lane & 15;
    const int half = lane >> 4;

    v8f accLo = {};
    v8f accHi = {};
#pragma unroll
    for (int k0 = 0; k0 < DIM; k0 += 4) {
        const int kb = k0 + half * 2;
        // A 16x4 f32: lanes 0-15 hold K=k0,k0+1 ; lanes 16-31 hold K=k0+2,k0+3
        v2f a = *(const v2f*)(h + (size_t)(rowBase + col) * DIM + kb);
        // B 4x16 f32: VGPR0 = rows K=kb ; VGPR1 = rows K=kb+1 (per half)
        v2f bl, bh;
        bl[0] = W[kb*DIM + col];        bl[1] = W[(kb+1)*DIM + col];
        bh[0] = W[kb*DIM + 16 + col];   bh[1] = W[(kb+1)*DIM + 16 + col];
        accLo = __builtin_amdgcn_wmma_f32_16x16x4_f32(false, a, false, bl, (short)0, accLo, false, false);
        accHi = __builtin_amdgcn_wmma_f32_16x16x4_f32(false, a, false, bh, (short)0, accHi, false, false);
    }
#pragma unroll
    for (int r = 0; r < 8; ++r) {
        int ro = rowBase + r + half * 8;        // D: VGPR r -> M=r (lanes 0-15), M=r+8 (lanes 16-31)
        hW[(size_t)ro*DIM + col]      = accLo[r];
        hW[(size_t)ro*DIM + 16 + col] = accHi[r];
    }
}

// ---------------- Kernel 4: edge loop (VALU-bound silu gating) ----------------
// wave = one src node, lane = channel k; per-lane accumulators, w_out applied once at the end.
__global__ void __launch_bounds__(256)
k_edge(const float* __restrict__ pt_ws,       // [N][4]
       const float* __restrict__ obj_ws,      // [N][4]
       const float* __restrict__ hW,          // [N][32]
       const float* __restrict__ w_dist,
       const float* __restrict__ w_out,
       const unsigned char* __restrict__ mask_ads,
       float* __restrict__ pernode,
       float* __restrict__ out)               // d_out: [0]=loss, [1..]=pos_relaxed_recon
{
    __shared__ float s_pos[NPG * 4];          // 4 KB
    __shared__ float s_hW[NPG * DIM];         // 32 KB (lane-contiguous: conflict-free)

    const int g    = blockIdx.x >> 5;         // 32 blocks per graph
    const int tid  = threadIdx.x;
    const int base = g * NPG;

    ((float4*)s_pos)[tid] = ((const float4*)pt_ws)[base + tid];
    const float4* src4 = (const float4*)(hW + (size_t)base * DIM);
#pragma unroll
    for (int it = 0; it < 8; ++it)
        ((float4*)s_hW)[tid + it*256] = src4[tid + it*256];
    __syncthreads();

    const int lane = tid & 31;
    const int wave = tid >> 5;
    const int srcLocal = ((blockIdx.x & 31) << 3) + wave;
    const int i = base + srcLocal;

    const float hw_s = s_hW[srcLocal*DIM + lane];
    const float wd   = w_dist[lane];
    const float wo   = w_out[lane];
    const float4 ps  = ((const float4*)s_pos)[srcLocal];
    float ax = 0.f, ay = 0.f, az = 0.f;

    for (int dst = 0; dst < NPG; ++dst) {
        if (dst == srcLocal) continue;                       // wave-uniform branch
        float4 pd = ((const float4*)s_pos)[dst];             // broadcast b128
        float rx = pd.x - ps.x, ry = pd.y - ps.y, rz = pd.z - ps.z;
        rx -= floorf(rx + 0.5f); ry -= floorf(ry + 0.5f); rz -= floorf(rz + 0.5f);
        float d2 = rx*rx + ry*ry + rz*rz;
        float t  = hw_s + s_hW[dst*DIM + lane] + d2*wd;      // (hW_s + hW_d) + d2*w_dist
        float m  = t * __builtin_amdgcn_rcpf(1.0f + __expf(-t));   // silu
        ax = fmaf(m, rx, ax);
        ay = fmaf(m, ry, ay);
        az = fmaf(m, rz, az);
    }

    // agg_c = sum_k w_out[k] * a_c[k]  (single cross-lane reduction per src)
    float vx = wo*ax, vy = wo*ay, vz = wo*az;
#pragma unroll
    for (int off = 16; off > 0; off >>= 1) {
        vx += __shfl_xor(vx, off, 32);
        vy += __shfl_xor(vy, off, 32);
        vz += __shfl_xor(vz, off, 32);
    }

    if (lane == 0) {
        float mk = mask_ads[i] ? 1.0f : 0.0f;
        float orx = vx*mk, ory = vy*mk, orz = vz*mk;
        float4 ob = ((const float4*)obj_ws)[i];
        pernode[i] = fabsf(ob.x - orx) + fabsf(ob.y - ory) + fabsf(ob.z - orz);
        out[1 + 3*i + 0] = ps.x - orx;
        out[1 + 3*i + 1] = ps.y - ory;
        out[1 + 3*i + 2] = ps.z - orz;
    }
}

// ---------------- Kernel 5: deterministic loss reduction ----------------
__global__ void k_reduce(const float* __restrict__ pernode, float* __restrict__ out, int N)
{
    __shared__ float s[256];
    float local = 0.f;
    for (int i = threadIdx.x; i < N; i += 256) local += pernode[i];  // fixed order
    s[threadIdx.x] = local;
    __syncthreads();
    for (int off = 128; off > 0; off >>= 1) {
        if (threadIdx.x < off) s[threadIdx.x] += s[threadIdx.x + off];
        __syncthreads();
    }
    if (threadIdx.x == 0) out[0] = s[0] / (float)(N * 3);
}

extern "C" void kernel_launch(void* const* d_in, const int* in_sizes, int n_in,
                              void* d_out, int out_size, void* d_ws, size_t ws_size,
                              hipStream_t stream)
{
    const float* pos            = (const float*)d_in[0];
    const float* pos_relaxed    = (const float*)d_in[1];
    const float* cell           = (const float*)d_in[2];
    const float* noise          = (const float*)d_in[3];
    const float* m_sched        = (const float*)d_in[4];
    const float* var_sched      = (const float*)d_in[5];
    const float* emb_z          = (const float*)d_in[6];
    const float* emb_t          = (const float*)d_in[7];
    const float* W_msg          = (const float*)d_in[8];
    const float* w_dist         = (const float*)d_in[9];
    const float* w_out          = (const float*)d_in[10];
    const int*   atomic_numbers = (const int*)d_in[11];
    const int*   t_graph        = (const int*)d_in[13];
    const unsigned char* mask_ads = (const unsigned char*)d_in[14];

    const int N = in_sizes[0] / 3;     // 8192
    const int G = in_sizes[2] / 9;     // 32   (n = N/G = 256, d = 32)

    float* ws      = (float*)d_ws;
    float* h_buf   = ws;                                  // N*32
    float* hW_buf  = ws + (size_t)N * 32;                 // N*32
    float* pt_ws   = ws + (size_t)N * 64;                 // N*4
    float* obj_ws  = ws + (size_t)N * 64 + (size_t)N * 4; // N*4
    float* pernode = ws + (size_t)N * 64 + (size_t)N * 8; // N
    float* out     = (float*)d_out;

    k_setup<<<G, NPG, 0, stream>>>(pos, pos_relaxed, cell, noise,
                                   m_sched, var_sched, t_graph, pt_ws, obj_ws);
    k_h<<<(N * DIM) / 256, 256, 0, stream>>>(emb_z, emb_t, atomic_numbers, t_graph,
                                             h_buf, N * DIM);
    k_hW_wmma<<<N / (16 * 8), 256, 0, stream>>>(h_buf, W_msg, hW_buf);
    k_edge<<<N / 8, 256, 0, stream>>>(pt_ws, obj_ws, hW_buf, w_dist, w_out,
                                      mask_ads, pernode, out);
    k_reduce<<<1, 256, 0, stream>>>(pernode, out, N);
}